// GCN_10368051052900
// MI455X (gfx1250) — compile-verified
//
#include <hip/hip_runtime.h>
#include <math.h>

typedef float v2f __attribute__((ext_vector_type(2)));
typedef float v8f __attribute__((ext_vector_type(8)));

#define FEAT 128

// ---------- degree (integer atomics: deterministic) ----------
__global__ void gcn_deg_kernel(const int* __restrict__ dst, int* __restrict__ deg, int E) {
    int e = blockIdx.x * blockDim.x + threadIdx.x;
    if (e < E) atomicAdd(&deg[dst[e]], 1);
}

__global__ void gcn_dinv_kernel(const int* __restrict__ deg, float* __restrict__ dinv, int N) {
    int i = blockIdx.x * blockDim.x + threadIdx.x;
    if (i < N) dinv[i] = rsqrtf((float)deg[i] + 1.0f);   // +1 = self loop, always > 0
}

// ---------- WMMA f32 GEMM: h1[N,4] = x[N,128] @ W1[128,4] ----------
// One wave per 16-row tile. B (W1) zero-padded to 128x16 in LDS.
__global__ void gcn_gemm1_wmma(const float* __restrict__ x, const float* __restrict__ W1,
                               float* __restrict__ h1, int N, int ntiles) {
    __shared__ float Wp[FEAT * 16];
    int t = threadIdx.x;                       // blockDim = 128 (4 waves)
    for (int i = t; i < FEAT * 16; i += 128) {
        int k = i >> 4, n = i & 15;
        Wp[i] = (n < 4) ? W1[k * 4 + n] : 0.0f;
    }
    __syncthreads();

    int wave = t >> 5;
    int lane = t & 31;
    int tile = blockIdx.x * 4 + wave;
    if (tile >= ntiles) return;                // wave-uniform: EXEC stays all-ones for WMMA
    int row0 = tile * 16;
    int m  = lane & 15;                        // M (A) / N (B,C)
    int hi = lane >> 4;                        // K sub-select for A/B fragments

    int row = row0 + m; if (row >= N) row = N - 1;  // clamp (N % 16 == 0 normally)
    const float* arow = x + (long)row * FEAT + hi * 2;

    v8f c = {0.f, 0.f, 0.f, 0.f, 0.f, 0.f, 0.f, 0.f};
    #pragma unroll 8
    for (int k = 0; k < FEAT / 4; ++k) {
        // A 16x4 f32 fragment: lanes 0-15 -> K=4k,4k+1 ; lanes 16-31 -> K=4k+2,4k+3
        v2f a = *(const v2f*)(arow + k * 4);
        // B 4x16 f32 fragment: VGPR0 = row K=4k(+2), VGPR1 = row K=4k+1(+3), N = m
        int kk = k * 4 + hi * 2;
        v2f b;
        b.x = Wp[kk * 16 + m];
        b.y = Wp[(kk + 1) * 16 + m];
        c = __builtin_amdgcn_wmma_f32_16x16x4_f32(false, a, false, b, (short)0, c, false, false);
    }

    // D 16x16 f32: VGPR v -> (M=v, N=lane) for lanes 0-15, (M=v+8, N=lane-16) for 16-31.
    // Only columns 0..3 are real outputs.
    if (m < 4) {
        int rbase = row0 + hi * 8;
        #pragma unroll
        for (int v = 0; v < 8; ++v) {
            int r = rbase + v;
            if (r < N) h1[(long)r * 4 + m] = c[v];
        }
    }
}

// ---------- per-edge normalized scatter-add (L2-resident, native f32 atomics) ----------
template <int F>
__global__ void gcn_agg_kernel(const int* __restrict__ src, const int* __restrict__ dst,
                               const float* __restrict__ dinv, const float* __restrict__ h,
                               float* __restrict__ outb, int E) {
    int e = blockIdx.x * blockDim.x + threadIdx.x;
    if (e >= E) return;
    int s = src[e], d = dst[e];
    float w = dinv[s] * dinv[d];
    #pragma unroll
    for (int f = 0; f < F; ++f)
        unsafeAtomicAdd(&outb[(long)d * F + f], h[(long)s * F + f] * w);
}

// ---------- fused epilogue: self-loop + bias + tanh + next tiny linear ----------
template <int FIN, int FOUT>
__global__ void gcn_finalize(const float* __restrict__ agg, const float* __restrict__ hraw,
                             const float* __restrict__ dinv, const float* __restrict__ b,
                             const float* __restrict__ Wnext, float* __restrict__ hnext, int N) {
    int i = blockIdx.x * blockDim.x + threadIdx.x;
    if (i >= N) return;
    float di = dinv[i];
    float sw = di * di;                       // self-loop norm = dinv^2
    float tv[FIN];
    #pragma unroll
    for (int f = 0; f < FIN; ++f)
        tv[f] = tanhf(agg[(long)i * FIN + f] + hraw[(long)i * FIN + f] * sw + b[f]);
    #pragma unroll
    for (int j = 0; j < FOUT; ++j) {
        float acc = 0.f;
        #pragma unroll
        for (int f = 0; f < FIN; ++f) acc += tv[f] * Wnext[f * FOUT + j];
        hnext[(long)i * FOUT + j] = acc;
    }
}

// Last layer: write tuple outputs (out = t3 @ Wc + bc  [N,16];  h = t3 [N,2])
__global__ void gcn_finalize3(const float* __restrict__ agg, const float* __restrict__ hraw,
                              const float* __restrict__ dinv, const float* __restrict__ b3,
                              const float* __restrict__ Wc, const float* __restrict__ bc,
                              float* __restrict__ out, float* __restrict__ hout, int N) {
    int i = blockIdx.x * blockDim.x + threadIdx.x;
    if (i >= N) return;
    float di = dinv[i];
    float sw = di * di;
    float t0 = tanhf(agg[(long)i * 2 + 0] + hraw[(long)i * 2 + 0] * sw + b3[0]);
    float t1 = tanhf(agg[(long)i * 2 + 1] + hraw[(long)i * 2 + 1] * sw + b3[1]);
    hout[(long)i * 2 + 0] = t0;
    hout[(long)i * 2 + 1] = t1;
    #pragma unroll
    for (int c = 0; c < 16; ++c)
        out[(long)i * 16 + c] = t0 * Wc[c] + t1 * Wc[16 + c] + bc[c];
}

extern "C" void kernel_launch(void* const* d_in, const int* in_sizes, int n_in,
                              void* d_out, int out_size, void* d_ws, size_t ws_size,
                              hipStream_t stream) {
    const float* x  = (const float*)d_in[0];
    const float* W1 = (const float*)d_in[1];
    const float* b1 = (const float*)d_in[2];
    const float* W2 = (const float*)d_in[3];
    const float* b2 = (const float*)d_in[4];
    const float* W3 = (const float*)d_in[5];
    const float* b3 = (const float*)d_in[6];
    const float* Wc = (const float*)d_in[7];
    const float* bc = (const float*)d_in[8];
    const int*   ei = (const int*)d_in[9];

    int N = in_sizes[0] / FEAT;     // 200000
    int E = in_sizes[9] / 2;        // 6400000
    const int* src = ei;
    const int* dst = ei + E;

    // Workspace layout (floats): deg | agg1 | agg2 | agg3 | dinv | h1 | h2 | h3
    float* ws = (float*)d_ws;
    long Nl = N;
    int*   deg  = (int*)ws;               // [N]
    float* agg1 = ws + 1 * Nl;            // [4N]
    float* agg2 = ws + 5 * Nl;            // [4N]
    float* agg3 = ws + 9 * Nl;            // [2N]
    float* dinv = ws + 11 * Nl;           // [N]
    float* h1   = ws + 12 * Nl;           // [4N]
    float* h2   = ws + 16 * Nl;           // [4N]
    float* h3   = ws + 20 * Nl;           // [2N]

    float* out  = (float*)d_out;          // [N,16]
    float* hout = out + (long)N * 16;     // [N,2]

    // Zero accumulators (deg + agg1 + agg2 + agg3 are contiguous: 11N floats)
    hipMemsetAsync(ws, 0, (size_t)(11 * Nl) * sizeof(float), stream);

    int eb = (E + 255) / 256;
    int nb = (N + 255) / 256;
    int ntiles = (N + 15) / 16;

    gcn_deg_kernel<<<eb, 256, 0, stream>>>(dst, deg, E);
    gcn_dinv_kernel<<<nb, 256, 0, stream>>>(deg, dinv, N);

    gcn_gemm1_wmma<<<(ntiles + 3) / 4, 128, 0, stream>>>(x, W1, h1, N, ntiles);

    gcn_agg_kernel<4><<<eb, 256, 0, stream>>>(src, dst, dinv, h1, agg1, E);
    gcn_finalize<4, 4><<<nb, 256, 0, stream>>>(agg1, h1, dinv, b1, W2, h2, N);

    gcn_agg_kernel<4><<<eb, 256, 0, stream>>>(src, dst, dinv, h2, agg2, E);
    gcn_finalize<4, 2><<<nb, 256, 0, stream>>>(agg2, h2, dinv, b2, W3, h3, N);

    gcn_agg_kernel<2><<<eb, 256, 0, stream>>>(src, dst, dinv, h3, agg3, E);
    gcn_finalize3<<<nb, 256, 0, stream>>>(agg3, h3, dinv, b3, Wc, bc, out, hout, N);
}